// DSMoE_20693152432792
// MI455X (gfx1250) — compile-verified
//
#include <hip/hip_runtime.h>
#include <hip/hip_bf16.h>
#include <math.h>

// ---------------- problem constants (match reference) ----------------
#define E_EXP   64
#define H_DIM   2048
#define I_DIM   704
#define G_GRP   8
#define TOPK_GROUP 4
#define K_TOP   8
#define N_TOK   4096          // B*S = 2*2048
#define CAP     1024          // 2*N*K/E
#define SCALE_F 2.5f

// ---------------- vector types ----------------
typedef __attribute__((ext_vector_type(2))) float v2f;
typedef __attribute__((ext_vector_type(8))) float v8f;
typedef __attribute__((ext_vector_type(4))) float v4f;
typedef __attribute__((ext_vector_type(4))) unsigned int v4u;
typedef __attribute__((ext_vector_type(4))) int v4i;
typedef __attribute__((ext_vector_type(8))) int v8i;

#if defined(__has_builtin)
#if __has_builtin(__builtin_amdgcn_tensor_load_to_lds) && \
    __has_builtin(__builtin_amdgcn_s_wait_tensorcnt)
#define USE_TDM 1
#endif
#endif
#ifndef USE_TDM
#define USE_TDM 0
#endif

static __device__ __forceinline__ v8f wmma_f32_4(v2f a, v2f b, v8f c) {
  // D = A(16x4,f32) * B(4x16,f32) + C(16x16,f32)
  return __builtin_amdgcn_wmma_f32_16x16x4_f32(false, a, false, b, (short)0, c,
                                               false, false);
}

static __device__ __forceinline__ float silu_f(float v) {
  return v / (1.0f + __expf(-v));
}

// ---------------- kernel 0: zero output ----------------
__global__ void dsmoe_zero_kernel(float* __restrict__ p, long n) {
  long i = (long)blockIdx.x * blockDim.x + threadIdx.x;
  long stride = (long)gridDim.x * blockDim.x;
  for (; i < n; i += stride) p[i] = 0.0f;
}

// ---------------- kernel 1: router logits + sigmoid ----------------
// grid = N_TOK blocks, 64 threads (one expert per thread), x row staged in LDS
__global__ void dsmoe_router_kernel(const float* __restrict__ x,
                                    const float* __restrict__ Wr,
                                    float* __restrict__ scores) {
  extern __shared__ float xs[];           // H_DIM floats
  const int n = blockIdx.x;
  {
    const v4f* src = (const v4f*)(x + (size_t)n * H_DIM);
    v4f* dst = (v4f*)xs;
    for (int h = threadIdx.x; h < H_DIM / 4; h += blockDim.x) dst[h] = src[h];
  }
  __syncthreads();
  const int e = threadIdx.x;
  const v4f* w = (const v4f*)(Wr + (size_t)e * H_DIM);
  const v4f* xv = (const v4f*)xs;
  float acc = 0.0f;
#pragma unroll 4
  for (int h = 0; h < H_DIM / 4; ++h) {
    v4f a = xv[h], b = w[h];
    acc += a.x * b.x + a.y * b.y + a.z * b.z + a.w * b.w;
  }
  scores[(size_t)n * E_EXP + e] = 1.0f / (1.0f + __expf(-acc));
}

// ---------------- kernel 2: group-limited top-k routing ----------------
__global__ void dsmoe_topk_kernel(const float* __restrict__ scores,
                                  const float* __restrict__ e_bias,
                                  int* __restrict__ topk_idx,
                                  float* __restrict__ topk_w) {
  const int n = blockIdx.x * blockDim.x + threadIdx.x;
  if (n >= N_TOK) return;
  float sc[E_EXP];
  for (int e = 0; e < E_EXP; ++e)
    sc[e] = scores[(size_t)n * E_EXP + e] + e_bias[e];

  float gs[G_GRP];
  for (int g = 0; g < G_GRP; ++g) {
    float m1 = -3.4e38f, m2 = -3.4e38f;
    for (int j = 0; j < E_EXP / G_GRP; ++j) {
      float v = sc[g * (E_EXP / G_GRP) + j];
      if (v > m1) { m2 = m1; m1 = v; }
      else if (v > m2) { m2 = v; }
    }
    gs[g] = m1 + m2;
  }
  bool gsel[G_GRP];
  for (int g = 0; g < G_GRP; ++g) gsel[g] = false;
  for (int t = 0; t < TOPK_GROUP; ++t) {
    float best = -3.4e38f; int bi = 0;
    for (int g = 0; g < G_GRP; ++g)
      if (!gsel[g] && gs[g] > best) { best = gs[g]; bi = g; }
    gsel[bi] = true;
  }
  float masked[E_EXP];
  for (int e = 0; e < E_EXP; ++e)
    masked[e] = gsel[e / (E_EXP / G_GRP)] ? sc[e] : 0.0f;
  int idx[K_TOP];
  for (int k = 0; k < K_TOP; ++k) {
    float best = -3.4e38f; int bi = 0;
    for (int e = 0; e < E_EXP; ++e)
      if (masked[e] > best) { best = masked[e]; bi = e; }
    idx[k] = bi;
    masked[bi] = -3.4e38f;
  }
  float w[K_TOP], wsum = 0.0f;
  for (int k = 0; k < K_TOP; ++k) {
    w[k] = scores[(size_t)n * E_EXP + idx[k]];
    wsum += w[k];
  }
  const float inv = SCALE_F / (wsum + 1e-20f);
  for (int k = 0; k < K_TOP; ++k) {
    topk_idx[n * K_TOP + k] = idx[k];
    topk_w[n * K_TOP + k] = w[k] * inv;
  }
}

// ---------------- kernel 3: deterministic per-expert row lists ----------------
__global__ void dsmoe_build_lists_kernel(const int* __restrict__ topk_idx,
                                         const float* __restrict__ topk_w,
                                         int* __restrict__ rowsL,
                                         float* __restrict__ scalesL,
                                         int* __restrict__ counts) {
  const int e = blockIdx.x;
  const int lane = threadIdx.x;        // 0..31
  int cnt = 0;
  const unsigned below = (lane == 0) ? 0u : ((1u << lane) - 1u);
  for (int base = 0; base < N_TOK * K_TOP; base += 32) {
    const int i = base + lane;         // N*K is a multiple of 32
    const bool m = (topk_idx[i] == e);
    const unsigned mask = (unsigned)__builtin_amdgcn_ballot_w32(m);
    if (m) {
      const int p = cnt + __popc(mask & below);
      if (p < CAP) {
        rowsL[e * CAP + p] = i / K_TOP;
        scalesL[e * CAP + p] = topk_w[i];
      }
    }
    cnt += __popc(mask);
  }
  const int c = cnt < CAP ? cnt : CAP;
  if (lane == 0) counts[e] = c;
  for (int p = c + lane; p < CAP; p += 32) rowsL[e * CAP + p] = -1;
}

// ---------------- kernel 4: fused expert MLP ----------------
#define MT      32          // rows per block
#define KCHUNK  128         // H staged per LDS chunk
#define XPAD    132         // 132 % 64 == 4 -> conflict-free; == 128 + TDM pad 4
#define HPAD    706         // 706 % 64 == 2 -> conflict-free row access
#define TILES_I (I_DIM / 64)   // 11
#define TILES_H (H_DIM / 64)   // 32

#if USE_TDM
// Issue one TDM gather-mode load: 16 scattered rows of KCHUNK f32 each into
// LDS, with pad_interval=128 dwords / pad_amount=4 dwords => row pitch XPAD.
static __device__ __forceinline__ void tdm_gather16(unsigned lds_off,
                                                    const float* gbase,
                                                    const int* rowTok,
                                                    int rbase) {
  unsigned long long ga = (unsigned long long)(size_t)gbase;
  v4u g0;
  g0[0] = 0x80000001u;                     // count=1, gather_mode=1, idx16
  g0[1] = lds_off;                         // lds_addr
  g0[2] = (unsigned)(ga & 0xFFFFFFFFu);    // global_addr[31:0]
  g0[3] = (unsigned)((ga >> 32) & 0x01FFFFFFu) | (2u << 30);  // addr hi | type=2
  v8i g1;
  g1[0] = (2 << 16)          // data_size = 4B
        | (1 << 20)          // pad_enable
        | (6 << 22)          // pad_interval = 128 dwords
        | (3 << 25);         // pad_amount = 4 dwords
  g1[1] = (H_DIM & 0xFFFF) << 16;                   // tensor_dim0 lo16
  g1[2] = (H_DIM >> 16) | ((N_TOK & 0xFFFF) << 16); // td0 hi | td1 lo
  g1[3] = (N_TOK >> 16) | (KCHUNK << 16);           // td1 hi | tile_dim0
  g1[4] = 16;                                       // tile_dim1 = #indices
  g1[5] = H_DIM;                                    // td0_stride lo32
  g1[6] = 0;                                        // td0_stride hi | td1_str lo
  g1[7] = 0;
  v4i g2, g3;
#pragma unroll
  for (int j = 0; j < 4; ++j) {
    int a0 = rowTok[rbase + 2 * j];     if (a0 < 0) a0 = 0;
    int a1 = rowTok[rbase + 2 * j + 1]; if (a1 < 0) a1 = 0;
    g2[j] = (a0 & 0xFFFF) | (a1 << 16);
    int b0 = rowTok[rbase + 8 + 2 * j];     if (b0 < 0) b0 = 0;
    int b1 = rowTok[rbase + 8 + 2 * j + 1]; if (b1 < 0) b1 = 0;
    g3[j] = (b0 & 0xFFFF) | (b1 << 16);
  }
  v8i gpad = 0;   // 6-arg toolchain form: extra descriptor group, zero-filled
  __builtin_amdgcn_tensor_load_to_lds(g0, g1, g2, g3, gpad, 0);
}
#endif

__global__ void dsmoe_fused_mlp_kernel(const float* __restrict__ x,
                                       const float* __restrict__ Wg,
                                       const float* __restrict__ Wu,
                                       const float* __restrict__ Wd,
                                       const int* __restrict__ rowsL,
                                       const float* __restrict__ scalesL,
                                       const int* __restrict__ counts,
                                       float* __restrict__ out,
                                       int tilesPerExpert) {
  extern __shared__ float smem[];
  float* xs = smem;                        // [MT][XPAD]
  float* hs = smem + MT * XPAD;            // [MT][HPAD]
  int*   rowTok   = (int*)(hs + MT * HPAD);   // [MT]
  float* rowScale = (float*)(rowTok + MT);    // [MT]

  const int tid  = threadIdx.x;
  const int wave = tid >> 5;
  const int lane = tid & 31;
  const int half = lane >> 4;              // selects K/M upper halves
  const int l    = lane & 15;

  int e, tile;
  if (rowsL) { e = blockIdx.x / tilesPerExpert; tile = blockIdx.x % tilesPerExpert; }
  else       { e = 0;                         tile = blockIdx.x; }
  const int rowBase = tile * MT;
  const int count = rowsL ? counts[e] : N_TOK;
  if (rowBase >= count) return;

  const float* wg = Wg + (size_t)e * H_DIM * I_DIM;
  const float* wu = Wu + (size_t)e * H_DIM * I_DIM;
  const float* wd = Wd + (size_t)e * I_DIM * H_DIM;

  if (tid < MT) {
    int tok = rowsL ? rowsL[e * CAP + rowBase + tid] : (rowBase + tid);
    rowTok[tid] = tok;
    rowScale[tid] = rowsL ? (tok >= 0 ? scalesL[e * CAP + rowBase + tid] : 0.0f)
                          : 1.0f;
  }
  __syncthreads();

  // ---------------- Stage A: gate & up GEMMs + SiLU ----------------
  for (int ip = 0; ip < TILES_I; ++ip) {
    const int colI = ip * 64 + wave * 16;
    v8f g0 = 0.0f, g1 = 0.0f, u0 = 0.0f, u1 = 0.0f;

    for (int kb = 0; kb < H_DIM; kb += KCHUNK) {
      __syncthreads();        // previous chunk fully consumed
#if USE_TDM
      if (wave == 0) {
        const unsigned lds0 = (unsigned)(unsigned long long)(size_t)xs;
        tdm_gather16(lds0, x + kb, rowTok, 0);
        tdm_gather16(lds0 + 16u * XPAD * 4u, x + kb, rowTok, 16);
        __builtin_amdgcn_s_wait_tensorcnt(0);
      }
#else
      // cooperative vectorized load of x chunk [MT][KCHUNK]
      for (int t = tid; t < MT * (KCHUNK / 4); t += 128) {
        const int r = t >> 5;                 // t / (KCHUNK/4)
        const int k4 = t & (KCHUNK / 4 - 1);
        const int tok = rowTok[r];
        v4f v = 0.0f;
        if (tok >= 0)
          v = *(const v4f*)(x + (size_t)tok * H_DIM + kb + 4 * k4);
        *(v4f*)(xs + r * XPAD + 4 * k4) = v;
      }
#endif
      __syncthreads();

      for (int k = 0; k < KCHUNK; k += 4) {
        // A fragments: lane(l,half) holds A[M=l][K=half*2 + {0,1}]
        v2f a0, a1;
        const float* xr0 = &xs[l * XPAD + k + half * 2];
        a0.x = xr0[0]; a0.y = xr0[1];
        const float* xr1 = &xs[(16 + l) * XPAD + k + half * 2];
        a1.x = xr1[0]; a1.y = xr1[1];
        // B fragments: lane(l,half) holds B[K=half*2 + {0,1}][N=l]
        const float* bg = wg + (size_t)(kb + k + half * 2) * I_DIM + colI + l;
        v2f bgv; bgv.x = bg[0]; bgv.y = bg[I_DIM];
        const float* bu = wu + (size_t)(kb + k + half * 2) * I_DIM + colI + l;
        v2f buv; buv.x = bu[0]; buv.y = bu[I_DIM];
        if (k == 0) {  // prefetch next weight chunk (global_prefetch_b8)
          __builtin_prefetch(bg + (size_t)KCHUNK * I_DIM, 0, 1);
          __builtin_prefetch(bu + (size_t)KCHUNK * I_DIM, 0, 1);
        }
        g0 = wmma_f32_4(a0, bgv, g0);
        g1 = wmma_f32_4(a1, bgv, g1);
        u0 = wmma_f32_4(a0, buv, u0);
        u1 = wmma_f32_4(a1, buv, u1);
      }
    }
    // h = silu(g) * u -> LDS.  C layout: vgpr j -> M = j + half*8, N = l.
#pragma unroll
    for (int j = 0; j < 8; ++j) {
      const int r0 = j + half * 8;
      hs[r0 * HPAD + colI + l]        = silu_f(g0[j]) * u0[j];
      hs[(16 + r0) * HPAD + colI + l] = silu_f(g1[j]) * u1[j];
    }
  }
  __syncthreads();

  // ---------------- Stage B: down GEMM + scatter-combine ----------------
  for (int hp = 0; hp < TILES_H; ++hp) {
    const int colH = hp * 64 + wave * 16;
    v8f y0 = 0.0f, y1 = 0.0f;
    for (int k = 0; k < I_DIM; k += 4) {
      v2f a0, a1;
      const float* hr0 = &hs[l * HPAD + k + half * 2];
      a0.x = hr0[0]; a0.y = hr0[1];
      const float* hr1 = &hs[(16 + l) * HPAD + k + half * 2];
      a1.x = hr1[0]; a1.y = hr1[1];
      const float* bd = wd + (size_t)(k + half * 2) * H_DIM + colH + l;
      v2f bdv; bdv.x = bd[0]; bdv.y = bd[H_DIM];
      y0 = wmma_f32_4(a0, bdv, y0);
      y1 = wmma_f32_4(a1, bdv, y1);
    }
#pragma unroll
    for (int j = 0; j < 8; ++j) {
      const int r0 = j + half * 8;
      const int tok0 = rowTok[r0];
      if (tok0 >= 0)
        atomicAdd(&out[(size_t)tok0 * H_DIM + colH + l], y0[j] * rowScale[r0]);
      const int r1 = 16 + r0;
      const int tok1 = rowTok[r1];
      if (tok1 >= 0)
        atomicAdd(&out[(size_t)tok1 * H_DIM + colH + l], y1[j] * rowScale[r1]);
    }
  }
}

// ---------------- host launch ----------------
extern "C" void kernel_launch(void* const* d_in, const int* in_sizes, int n_in,
                              void* d_out, int out_size, void* d_ws, size_t ws_size,
                              hipStream_t stream) {
  const float* x       = (const float*)d_in[0];  // [B,S,H] -> [N,H]
  const float* Wr      = (const float*)d_in[1];  // [E,H]
  const float* e_bias  = (const float*)d_in[2];  // [E]
  const float* w_gate  = (const float*)d_in[3];  // [E,H,I]
  const float* w_up    = (const float*)d_in[4];  // [E,H,I]
  const float* w_down  = (const float*)d_in[5];  // [E,I,H]
  const float* sh_gate = (const float*)d_in[6];  // [H,I]
  const float* sh_up   = (const float*)d_in[7];  // [H,I]
  const float* sh_down = (const float*)d_in[8];  // [I,H]
  float* out = (float*)d_out;

  // workspace partition (< 2 MB total)
  char* ws = (char*)d_ws;
  float* scores  = (float*)ws;                       ws += (size_t)N_TOK * E_EXP * 4;
  int*   tk_idx  = (int*)ws;                         ws += (size_t)N_TOK * K_TOP * 4;
  float* tk_w    = (float*)ws;                       ws += (size_t)N_TOK * K_TOP * 4;
  int*   rowsL   = (int*)ws;                         ws += (size_t)E_EXP * CAP * 4;
  float* scalesL = (float*)ws;                       ws += (size_t)E_EXP * CAP * 4;
  int*   counts  = (int*)ws;

  dsmoe_zero_kernel<<<2048, 256, 0, stream>>>(out, (long)N_TOK * H_DIM);
  dsmoe_router_kernel<<<N_TOK, 64, H_DIM * sizeof(float), stream>>>(x, Wr, scores);
  dsmoe_topk_kernel<<<N_TOK / 64, 64, 0, stream>>>(scores, e_bias, tk_idx, tk_w);
  dsmoe_build_lists_kernel<<<E_EXP, 32, 0, stream>>>(tk_idx, tk_w, rowsL, scalesL, counts);

  const size_t lds = (size_t)(MT * XPAD + MT * HPAD) * 4 + MT * 4 + MT * 4;
  // routed experts: E * (CAP/MT) tiles
  dsmoe_fused_mlp_kernel<<<E_EXP * (CAP / MT), 128, lds, stream>>>(
      x, w_gate, w_up, w_down, rowsL, scalesL, counts, out, CAP / MT);
  // shared expert: identity row mapping, scale 1
  dsmoe_fused_mlp_kernel<<<N_TOK / MT, 128, lds, stream>>>(
      x, sh_gate, sh_up, sh_down, nullptr, nullptr, nullptr, out, N_TOK / MT);
}